// graph_embedding_Net_22247930594058
// MI455X (gfx1250) — compile-verified
//
#include <hip/hip_runtime.h>
#include <hip/hip_bf16.h>

#define Bdim 8
#define Ndim 512
#define Pdim 32
#define Titer 4

typedef __attribute__((ext_vector_type(2))) float v2f;
typedef __attribute__((ext_vector_type(8))) float v8f;

// ---------------------------------------------------------------------------
// Stage A: t[b,i,p] = blin[p] + sum_q wlin[p,q] * acc_q
//          acc_q   = (1/n) * sum_j relu(Esrc[b,j,i]*wv[q]+bv[q]) * adj[b,i,j]
// One wave per (b,i); lane = p. relu is inside the j-sum -> fused VALU path.
// ---------------------------------------------------------------------------
__global__ __launch_bounds__(256) void edge_pre(
    const float* __restrict__ Esrc,   // [B,N,N] accessed transposed: E[b,j,i]
    const float* __restrict__ adj,    // [B,N,N] adj[b,i,j]
    const float* __restrict__ wv,     // [P,1]
    const float* __restrict__ bv,     // [P]
    const float* __restrict__ wlin,   // [P,P]
    const float* __restrict__ blin,   // [P]
    float* __restrict__ outp)         // [B,N,P]
{
    const int lane = threadIdx.x & 31;
    const int wave = threadIdx.x >> 5;
    const int gw   = blockIdx.x * 8 + wave;     // 0 .. B*N-1
    const int b    = gw >> 9;                   // /N
    const int i    = gw & (Ndim - 1);

    const float wp = wv[lane];
    const float bp = bv[lane];
    const float* Ebase = Esrc + (size_t)b * Ndim * Ndim;            // [j*N + i]
    const float* Arow  = adj  + ((size_t)b * Ndim + i) * Ndim;      // [j]

    float acc = 0.f;
    for (int j0 = 0; j0 < Ndim; j0 += 32) {
        float ecol = Ebase[(size_t)(j0 + lane) * Ndim + i];
        float arow = Arow[j0 + lane];
#pragma unroll
        for (int jj = 0; jj < 32; ++jj) {
            float e = __shfl(ecol, jj, 32);
            float a = __shfl(arow, jj, 32);
            acc += fmaxf(fmaf(e, wp, bp), 0.f) * a;
        }
    }
    acc *= (1.0f / (float)Ndim);

    // small linear across the wave: gather acc_q with readlane-style shuffles
    float o = blin[lane];
    const float* wrow = wlin + lane * Pdim;
#pragma unroll
    for (int q = 0; q < Pdim; ++q)
        o = fmaf(__shfl(acc, q, 32), wrow[q], o);

    outp[((size_t)b * Ndim + i) * Pdim + lane] = o;
}

// ---------------------------------------------------------------------------
// t2lin[b,i,p] = b2[p] + sum_q Mu[b,i,q] * w2[p,q]   (one wave per row)
// ---------------------------------------------------------------------------
__global__ __launch_bounds__(256) void mu_lin(
    const float* __restrict__ Mu,     // [B,N,P]
    const float* __restrict__ w2,     // [P,P]
    const float* __restrict__ b2,     // [P]
    float* __restrict__ t2lin)        // [B,N,P]
{
    const int lane = threadIdx.x & 31;
    const int gw   = blockIdx.x * 8 + (threadIdx.x >> 5);   // 0 .. B*N-1
    const float mu = Mu[(size_t)gw * Pdim + lane];
    float o = b2[lane];
    const float* wrow = w2 + lane * Pdim;
#pragma unroll
    for (int q = 0; q < Pdim; ++q)
        o = fmaf(__shfl(mu, q, 32), wrow[q], o);
    t2lin[(size_t)gw * Pdim + lane] = o;
}

// ---------------------------------------------------------------------------
// MuOut[b,i,p] = relu( (t1 + (adjG @ t2lin)/n + t34) / 3 )
// Big per-batch GEMM [512x512]x[512x32] on the fp32 WMMA pipe:
// one wave = one 16x16 tile, K-loop of 128 V_WMMA_F32_16X16X4_F32.
// ---------------------------------------------------------------------------
__global__ __launch_bounds__(256) void mu_update_wmma(
    const float* __restrict__ adjG,   // [B,N,N]
    const float* __restrict__ t2lin,  // [B,N,P]
    const float* __restrict__ t1,     // [B,N,P]
    const float* __restrict__ t34,    // [B,N,P]
    float* __restrict__ MuOut)        // [B,N,P]
{
    const int lane = threadIdx.x & 31;
    const int wave = threadIdx.x >> 5;
    const int gw   = blockIdx.x * 8 + wave;   // 0 .. B*(N/16)*(P/16)-1 = 511
    const int b    = gw >> 6;                 // 64 tiles per batch
    const int r    = gw & 63;
    const int it   = r >> 1;                  // row tile 0..31
    const int jt   = r & 1;                   // col tile 0..1
    const int m_   = lane & 15;
    const int half = lane >> 4;
    const int i0   = it * 16;
    const int p0   = jt * 16;

    // A frag: lane holds A[i0+m_][k + 2*half + {0,1}]  (aligned float2)
    const float* Abase = adjG + ((size_t)b * Ndim + (i0 + m_)) * Ndim + half * 2;
    // B frag: lane holds B[k + 2*half + {0,1}][p0+m_]
    const float* Bbase = t2lin + (size_t)b * Ndim * Pdim
                       + (size_t)(half * 2) * Pdim + (p0 + m_);

    v8f c = {};
    for (int k = 0; k < Ndim; k += 4) {
        v2f a = *(const v2f*)(Abase + k);
        const float* bp = Bbase + (size_t)k * Pdim;
        v2f bb;
        bb.x = bp[0];
        bb.y = bp[Pdim];
        c = __builtin_amdgcn_wmma_f32_16x16x4_f32(
                /*neg_a=*/false, a, /*neg_b=*/false, bb,
                /*c_mod=*/(short)0, c, /*reuse_a=*/false, /*reuse_b=*/false);
    }

    const float invn = 1.0f / (float)Ndim;
    const int col = p0 + m_;
#pragma unroll
    for (int rr = 0; rr < 8; ++rr) {
        int m = rr + half * 8;                // C/D layout: VGPR rr <-> M
        size_t idx = ((size_t)b * Ndim + (i0 + m)) * Pdim + col;
        float v = (t1[idx] + c[rr] * invn + t34[idx]) * (1.0f / 3.0f);
        MuOut[idx] = fmaxf(v, 0.f);
    }
}

// ---------------------------------------------------------------------------
// Readout: one wave per batch. out[b,0,{0,1}]
// ---------------------------------------------------------------------------
__global__ __launch_bounds__(256) void readout(
    const float* __restrict__ Mu,     // [B,N,P]
    const int*   __restrict__ e,      // [B,2]
    const float* __restrict__ w6, const float* __restrict__ b6,
    const float* __restrict__ w7, const float* __restrict__ b7,
    const float* __restrict__ w5, const float* __restrict__ b5,
    float* __restrict__ out)          // [B,2]
{
    const int lane = threadIdx.x & 31;
    const int b    = threadIdx.x >> 5;        // 8 waves = 8 batches
    const float* Mub = Mu + (size_t)b * Ndim * Pdim;

    float s = 0.f;
    for (int i = 0; i < Ndim; ++i)
        s += Mub[(size_t)i * Pdim + lane];
    s *= (1.0f / (float)Ndim);

    const int e0 = e[b * 2 + 0];
    const int e1 = e[b * 2 + 1];
    float mt = 0.5f * (Mub[(size_t)e0 * Pdim + lane] + Mub[(size_t)e1 * Pdim + lane]);

    float t6 = b6[lane], t7 = b7[lane];
    const float* w6r = w6 + lane * Pdim;
    const float* w7r = w7 + lane * Pdim;
#pragma unroll
    for (int q = 0; q < Pdim; ++q) {
        t6 = fmaf(__shfl(s,  q, 32), w6r[q], t6);
        t7 = fmaf(__shfl(mt, q, 32), w7r[q], t7);
    }
    float r6 = fmaxf(t6, 0.f);
    float r7 = fmaxf(t7, 0.f);

    // w5: [2, 2P] ; out[b,c] = b5[c] + sum_p r6*w5[c,p] + r7*w5[c,P+p]
    float p0c = r6 * w5[lane]            + r7 * w5[Pdim + lane];
    float p1c = r6 * w5[2 * Pdim + lane] + r7 * w5[3 * Pdim + lane];
#pragma unroll
    for (int off = 16; off > 0; off >>= 1) {
        p0c += __shfl_xor(p0c, off, 32);
        p1c += __shfl_xor(p1c, off, 32);
    }
    if (lane == 0) {
        out[b * 2 + 0] = p0c + b5[0];
        out[b * 2 + 1] = p1c + b5[1];
    }
}

extern "C" void kernel_launch(void* const* d_in, const int* in_sizes, int n_in,
                              void* d_out, int out_size, void* d_ws, size_t ws_size,
                              hipStream_t stream) {
    (void)in_sizes; (void)n_in; (void)out_size; (void)ws_size;

    const float* E    = (const float*)d_in[0];
    const float* Mu_  = (const float*)d_in[1];
    const float* W    = (const float*)d_in[2];
    const float* adjF = (const float*)d_in[3];
    const float* adjG = (const float*)d_in[4];
    /* d_in[5] = T (fixed 4), d_in[7] = n (512), d_in[8] = bt_size (8) */
    const int*   ein  = (const int*)d_in[6];
    const float* w12  = (const float*)d_in[9];
    const float* b12  = (const float*)d_in[10];
    const float* w11  = (const float*)d_in[11];
    const float* b11  = (const float*)d_in[12];
    const float* w2   = (const float*)d_in[13];
    const float* b2   = (const float*)d_in[14];
    const float* w4   = (const float*)d_in[15];
    const float* b4   = (const float*)d_in[16];
    const float* w3   = (const float*)d_in[17];
    const float* b3   = (const float*)d_in[18];
    const float* w6   = (const float*)d_in[19];
    const float* b6   = (const float*)d_in[20];
    const float* w7   = (const float*)d_in[21];
    const float* b7   = (const float*)d_in[22];
    const float* w5   = (const float*)d_in[23];
    const float* b5   = (const float*)d_in[24];
    float* out = (float*)d_out;

    const size_t BNP = (size_t)Bdim * Ndim * Pdim;   // 131072 floats
    float* t1    = (float*)d_ws;
    float* t34   = t1 + BNP;
    float* t2lin = t34 + BNP;
    float* MuWs  = t2lin + BNP;                      // total 2 MB of ws

    dim3 blk(256);
    const int rowBlocks = (Bdim * Ndim) / 8;         // 8 waves (rows) per block

    // Stage A: fused edge transforms + weighted reductions + small linears
    edge_pre<<<rowBlocks, blk, 0, stream>>>(E, adjF, w12, b12, w11, b11, t1);
    edge_pre<<<rowBlocks, blk, 0, stream>>>(W, adjG, w4, b4, w3, b3, t34);

    // Stage B: T iterations of message passing (WMMA GEMM + fused update)
    for (int t = 0; t < Titer; ++t) {
        const float* MuIn = (t == 0) ? Mu_ : MuWs;
        mu_lin<<<rowBlocks, blk, 0, stream>>>(MuIn, w2, b2, t2lin);
        mu_update_wmma<<<(Bdim * 64) / 8, blk, 0, stream>>>(adjG, t2lin, t1, t34, MuWs);
    }

    // Stage C: readout
    readout<<<1, blk, 0, stream>>>(MuWs, ein, w6, b6, w7, b7, w5, b5, out);
}